// GeometricFeatureExtractor_25142738550967
// MI455X (gfx1250) — compile-verified
//
#include <hip/hip_runtime.h>
#include <math.h>

typedef __attribute__((ext_vector_type(2))) float v2f;
typedef __attribute__((ext_vector_type(8))) float v8f;

#define BATCH 16
#define NPTS  2048
#define NTIL  (NPTS / 16)   // 128 16-wide tiles per dimension

// ---------------- small float3 helpers ----------------
__device__ __forceinline__ float3 ldf3(const float* p) { return make_float3(p[0], p[1], p[2]); }
__device__ __forceinline__ float3 f3sub(float3 a, float3 b) { return make_float3(a.x-b.x, a.y-b.y, a.z-b.z); }
__device__ __forceinline__ float  f3dot(float3 a, float3 b) { return a.x*b.x + a.y*b.y + a.z*b.z; }
__device__ __forceinline__ float3 f3scale(float3 a, float s){ return make_float3(a.x*s, a.y*s, a.z*s); }
__device__ __forceinline__ float3 f3norm(float3 v) {
    float n   = sqrtf(f3dot(v, v));
    float inv = 1.0f / fmaxf(n, 1e-12f);
    return f3scale(v, inv);
}
__device__ __forceinline__ float3 f3cross(float3 a, float3 b) {
    return make_float3(a.y*b.z - a.z*b.y, a.z*b.x - a.x*b.z, a.x*b.y - a.y*b.x);
}

// ---------------- Kernel 1: angles / origins / frames (tiny, elementwise) ----------------
__global__ void __launch_bounds__(256)
geom_feats_kernel(const float* __restrict__ coords,
                  float* __restrict__ ang,     // (B, N, 3)
                  float* __restrict__ org,     // (B, N-2, 3)
                  float* __restrict__ frm) {   // (B, N-2, 3, 3)
    int idx = blockIdx.x * blockDim.x + threadIdx.x;
    if (idx >= BATCH * NPTS) return;
    int b = idx / NPTS;
    int i = idx - b * NPTS;
    const float* C = coords + (size_t)b * NPTS * 3;

    // ---- angles ----
    float a0 = 0.0f, a1 = 0.0f;
    if (i >= 1 && i < NPTS - 2) {
        float3 cm = ldf3(C + (size_t)(i - 1) * 3);
        float3 c0 = ldf3(C + (size_t)(i    ) * 3);
        float3 c1 = ldf3(C + (size_t)(i + 1) * 3);
        float3 c2 = ldf3(C + (size_t)(i + 2) * 3);
        float3 dm = f3norm(f3sub(c0, cm));   // d_{i-1}
        float3 d0 = f3norm(f3sub(c1, c0));   // d_i
        float3 dp = f3norm(f3sub(c2, c1));   // d_{i+1}
        a0 = acosf(fminf(fmaxf(f3dot(dm, d0), -1.0f), 1.0f));  // ang[i-1]
        a1 = acosf(fminf(fmaxf(f3dot(d0, dp), -1.0f), 1.0f));  // ang[i]
    }
    float* ap = ang + (size_t)idx * 3;
    ap[0] = a0; ap[1] = a1; ap[2] = 0.0f;

    // ---- origins + frames ----
    if (i < NPTS - 2) {
        float3 c0 = ldf3(C + (size_t)(i    ) * 3);
        float3 c1 = ldf3(C + (size_t)(i + 1) * 3);
        float3 c2 = ldf3(C + (size_t)(i + 2) * 3);
        size_t oi = (size_t)b * (NPTS - 2) + i;

        float* op = org + oi * 3;
        op[0] = c0.x; op[1] = c0.y; op[2] = c0.z;

        float3 x = f3norm(f3sub(c1, c0));
        float3 v = f3sub(c2, c0);
        float  p = f3dot(v, x);
        float3 y = f3norm(make_float3(v.x - p*x.x, v.y - p*x.y, v.z - p*x.z));
        float3 z = f3cross(x, y);

        // frames[b,i,r,c]: columns are x,y,z
        float* fp = frm + oi * 9;
        fp[0] = x.x; fp[1] = y.x; fp[2] = z.x;
        fp[3] = x.y; fp[4] = y.y; fp[5] = z.y;
        fp[6] = x.z; fp[7] = y.z; fp[8] = z.z;
    }
}

// ---------------- Kernel 2: pairwise distances via V_WMMA_F32_16X16X4_F32 ----------------
// One block = 8 waves handles a 16-row strip of one batch's 2048x2048 matrix.
// Each wave computes 16 column tiles (one WMMA per 16x16 tile, K=4 with z-pad 0).
__global__ void __launch_bounds__(256)
dist_wmma_kernel(const float* __restrict__ coords, float* __restrict__ dist) {
    const int lane = threadIdx.x & 31;
    const int wave = threadIdx.x >> 5;        // 0..7
    const int lm   = lane & 15;
    const bool hi  = lane >= 16;

    const int blk = blockIdx.x;               // b * NTIL + mt
    const int b   = blk / NTIL;
    const int mt  = blk - b * NTIL;
    const int m0  = mt * 16;

    const float* cb = coords + (size_t)b * NPTS * 3;

    // A tile: 16x4 (rows m0..m0+15, K = x,y,z,0). Lane L and L+16 load the same row.
    const float* apn = cb + (size_t)(m0 + lm) * 3;
    float ax = apn[0], ay = apn[1], az = apn[2];
    v2f a;
    a.x = hi ? az   : ax;
    a.y = hi ? 0.0f : ay;
    float sqA = ax*ax + ay*ay + az*az;        // |row (m0+lm)|^2, known by every lane

    // Gather the 8 row-norms this lane's D registers correspond to.
    float sm[8];
#pragma unroll
    for (int v = 0; v < 8; ++v)
        sm[v] = __shfl(sqA, (hi ? 8 : 0) + v, 32);

    float* dmat = dist + (size_t)b * NPTS * NPTS;

    for (int t = 0; t < 16; ++t) {
        const int n0 = (wave * 16 + t) * 16;

        // B tile: 4x16 (cols n0..n0+15). Same per-lane point as A-pattern.
        const float* bpn = cb + (size_t)(n0 + lm) * 3;
        float bx = bpn[0], by = bpn[1], bz = bpn[2];
        v2f bv;
        bv.x = hi ? bz   : bx;
        bv.y = hi ? 0.0f : by;
        float sqB = bx*bx + by*by + bz*bz;    // |col (n0+lm)|^2

        v8f c = {};
        c = __builtin_amdgcn_wmma_f32_16x16x4_f32(
            /*neg_a=*/false, a, /*neg_b=*/false, bv,
            /*c_mod=*/(short)0, c, /*reuse_a=*/false, /*reuse_b=*/false);

        // D layout: VGPR v -> row m0 + v + (hi?8:0), col n0 + lm
#pragma unroll
        for (int v = 0; v < 8; ++v) {
            float d2 = sm[v] + sqB - 2.0f * c[v];
            // raw v_sqrt_f32 (~1 ULP) — skip libm's denorm/fixup expansion
            float dd = __builtin_amdgcn_sqrtf(fmaxf(d2, 0.0f));
            int m = m0 + (hi ? 8 : 0) + v;
            // 268 MB streaming output, never re-read: bypass cache retention (TH=NT)
            __builtin_nontemporal_store(dd, &dmat[(size_t)m * NPTS + (n0 + lm)]);
        }
    }
}

extern "C" void kernel_launch(void* const* d_in, const int* in_sizes, int n_in,
                              void* d_out, int out_size, void* d_ws, size_t ws_size,
                              hipStream_t stream) {
    (void)in_sizes; (void)n_in; (void)out_size; (void)d_ws; (void)ws_size;
    const float* coords = (const float*)d_in[0];
    float* out = (float*)d_out;

    const size_t angOff = 0;
    const size_t orgOff = (size_t)BATCH * NPTS * 3;                       // 98304
    const size_t frmOff = orgOff + (size_t)BATCH * (NPTS - 2) * 3;        // 196512
    const size_t dstOff = frmOff + (size_t)BATCH * (NPTS - 2) * 9;        // 491136

    geom_feats_kernel<<<(BATCH * NPTS + 255) / 256, 256, 0, stream>>>(
        coords, out + angOff, out + orgOff, out + frmOff);

    dist_wmma_kernel<<<BATCH * NTIL, 256, 0, stream>>>(coords, out + dstOff);
}